// OrderRiderGNN_43791486550107
// MI455X (gfx1250) — compile-verified
//
#include <hip/hip_runtime.h>
#include <hip/hip_bf16.h>

typedef float v2f __attribute__((ext_vector_type(2)));
typedef float v8f __attribute__((ext_vector_type(8)));

// ---------------------------------------------------------------------------
// Monotonic float<->uint encoding so float max == unsigned max (exact).
// ---------------------------------------------------------------------------
__device__ __forceinline__ unsigned f32_key(float f) {
  unsigned b = __float_as_uint(f);
  return (b & 0x80000000u) ? ~b : (b | 0x80000000u);
}
__device__ __forceinline__ float key_f32(unsigned k) {
  unsigned b = (k & 0x80000000u) ? (k & 0x7fffffffu) : ~k;
  return __uint_as_float(b);
}
// key(-inf) = ~0xFF800000 = 0x007FFFFF
#define NEG_INF_KEY 0x007FFFFFu

// ---------------------------------------------------------------------------
// Dense f32 GEMM: C[M,N] = A[M,K] * B[K,N] using V_WMMA_F32_16X16X4_F32.
// One wave per 16x16 C tile. blockDim.x = 32*(N/16); gridDim.x = ceil(M/16).
// A 16x4 tile layout (ISA 7.12.2): lanes 0-15 -> M=0..15 / K={0,1},
//                                  lanes 16-31 -> M=0..15 / K={2,3}.
// B 4x16 tile: VGPR j, lane l: B[(l>>4)*2 + j, l&15].
// C tile: VGPR r, lanes 0-15 -> row r; lanes 16-31 -> row r+8.
// ---------------------------------------------------------------------------
__global__ void gat_gemm_f32_wmma(const float* __restrict__ A,
                                  const float* __restrict__ B,
                                  float* __restrict__ C,
                                  int M, int K, int N) {
  const int lane = threadIdx.x & 31;
  const int wave = threadIdx.x >> 5;
  const int m0 = blockIdx.x * 16;
  const int n0 = wave * 16;
  const int mn = lane & 15;            // row index (A) / col index (B,C)
  const int kb = (lane >> 4) << 1;     // 0 or 2

  int arow = m0 + mn;
  if (arow >= M) arow = M - 1;         // clamp; stores are guarded below

  const float* Arow = A + (size_t)arow * K;
  v8f acc = {};
#pragma unroll 4
  for (int k0 = 0; k0 < K; k0 += 4) {
    v2f a = *(const v2f*)(Arow + k0 + kb);          // 8B aligned (kb even, k0%4==0)
    v2f b;
    b[0] = B[(size_t)(k0 + kb) * N + n0 + mn];
    b[1] = B[(size_t)(k0 + kb + 1) * N + n0 + mn];
    acc = __builtin_amdgcn_wmma_f32_16x16x4_f32(
        /*neg_a=*/false, a, /*neg_b=*/false, b,
        /*c_mod=*/(short)0, acc, /*reuse_a=*/false, /*reuse_b=*/false);
  }

  const int rbase = (lane >> 4) << 3;  // +0 or +8
#pragma unroll
  for (int r = 0; r < 8; ++r) {
    int row = m0 + r + rbase;
    if (row < M) C[(size_t)row * N + n0 + mn] = acc[r];
  }
}

// ---------------------------------------------------------------------------
// Per-node attention logits: out_s[i] = h[i,:].a_src ; out_d[i] = h[i,:].a_dst
// One wave per node, float4 loads, wave32 shuffle reduction.
// ---------------------------------------------------------------------------
__global__ void gat_alphas(const float* __restrict__ H,
                           const float* __restrict__ a_s,
                           const float* __restrict__ a_d,
                           float* __restrict__ out_s,
                           float* __restrict__ out_d,
                           int Nnodes, int F) {
  const int lane = threadIdx.x & 31;
  const int node = (int)((blockIdx.x * (long long)blockDim.x + threadIdx.x) >> 5);
  if (node >= Nnodes) return;
  const float* row = H + (size_t)node * F;
  float ss = 0.f, sd = 0.f;
  for (int j = lane * 4; j < F; j += 128) {
    float4 h = *(const float4*)(row + j);
    float4 s4 = *(const float4*)(a_s + j);
    float4 d4 = *(const float4*)(a_d + j);
    ss += h.x * s4.x + h.y * s4.y + h.z * s4.z + h.w * s4.w;
    sd += h.x * d4.x + h.y * d4.y + h.z * d4.z + h.w * d4.w;
  }
#pragma unroll
  for (int off = 16; off; off >>= 1) {
    ss += __shfl_xor(ss, off, 32);
    sd += __shfl_xor(sd, off, 32);
  }
  if (lane == 0) { out_s[node] = ss; out_d[node] = sd; }
}

// ---------------------------------------------------------------------------
// Init: emax keys to key(-inf), denom to 0, aggregation buffer to 0.
// ---------------------------------------------------------------------------
__global__ void gat_init(unsigned* __restrict__ emaxkey,
                         float* __restrict__ denom,
                         float* __restrict__ agg,
                         int Nnodes, int F) {
  long long i = blockIdx.x * (long long)blockDim.x + threadIdx.x;
  long long total = (long long)Nnodes * F;
  if (i < total) agg[i] = 0.f;
  if (i < Nnodes) { emaxkey[i] = NEG_INF_KEY; denom[i] = 0.f; }
}

// ---------------------------------------------------------------------------
// Edge pass 1: e = LeakyReLU(as[src] + ad[dst]); segment-max into emaxkey[dst].
// Edges [0,E) from edge_index, [E, E+N) are self loops (matches reference).
// ---------------------------------------------------------------------------
__global__ void gat_edge_logits(const int* __restrict__ ei,
                                const float* __restrict__ as_,
                                const float* __restrict__ ad_,
                                float* __restrict__ ew,
                                unsigned* __restrict__ emaxkey,
                                int E, int Nnodes) {
  int e = blockIdx.x * blockDim.x + threadIdx.x;
  if (e >= E + Nnodes) return;
  int s, d;
  if (e < E) { s = ei[e]; d = ei[E + e]; } else { s = d = e - E; }
  float v = as_[s] + ad_[d];
  v = (v >= 0.f) ? v : 0.2f * v;
  ew[e] = v;
  atomicMax(&emaxkey[d], f32_key(v));
}

// ---------------------------------------------------------------------------
// Edge pass 2: w = exp(e - emax[dst]); segment-sum into denom[dst].
// Rewrites ew in place (logit -> weight).
// ---------------------------------------------------------------------------
__global__ void gat_edge_expsum(const int* __restrict__ ei,
                                float* __restrict__ ew,
                                const unsigned* __restrict__ emaxkey,
                                float* __restrict__ denom,
                                int E, int Nnodes) {
  int e = blockIdx.x * blockDim.x + threadIdx.x;
  if (e >= E + Nnodes) return;
  int d = (e < E) ? ei[E + e] : (e - E);
  float w = __expf(ew[e] - key_f32(emaxkey[d]));
  ew[e] = w;
  atomicAdd(&denom[d], w);
}

// ---------------------------------------------------------------------------
// Edge pass 3: agg[dst,:] += (w/denom[dst]) * h[src,:].
// One wave per edge; lane covers 4 features via float4 gather (coalesced
// 512B per wave), f32 atomics land in L2 (agg fits in 192MB L2).
// ---------------------------------------------------------------------------
__global__ void gat_edge_scatter(const int* __restrict__ ei,
                                 const float* __restrict__ ew,
                                 const float* __restrict__ denom,
                                 const float* __restrict__ H,
                                 float* __restrict__ agg,
                                 int E, int Nnodes, int F) {
  const int lane = threadIdx.x & 31;
  const int e = (int)((blockIdx.x * (long long)blockDim.x + threadIdx.x) >> 5);
  if (e >= E + Nnodes) return;
  int s, d;
  if (e < E) { s = ei[e]; d = ei[E + e]; } else { s = d = e - E; }
  float alpha = ew[e] / denom[d];
  const float* hs = H + (size_t)s * F;
  float* od = agg + (size_t)d * F;
  for (int j = lane * 4; j < F; j += 128) {
    float4 h = *(const float4*)(hs + j);
    atomicAdd(od + j + 0, alpha * h.x);
    atomicAdd(od + j + 1, alpha * h.y);
    atomicAdd(od + j + 2, alpha * h.z);
    atomicAdd(od + j + 3, alpha * h.w);
  }
}

// ---------------------------------------------------------------------------
// Finalize (in place): io += bias, optional ReLU.
// ---------------------------------------------------------------------------
__global__ void gat_finalize(float* __restrict__ io,
                             const float* __restrict__ bias,
                             int Nnodes, int F, int do_relu) {
  long long i = blockIdx.x * (long long)blockDim.x + threadIdx.x;
  if (i >= (long long)Nnodes * F) return;
  float v = io[i] + bias[(int)(i % F)];
  if (do_relu) v = fmaxf(v, 0.f);
  io[i] = v;
}

// ---------------------------------------------------------------------------
extern "C" void kernel_launch(void* const* d_in, const int* in_sizes, int n_in,
                              void* d_out, int out_size, void* d_ws, size_t ws_size,
                              hipStream_t stream) {
  const float* X      = (const float*)d_in[0];
  const int*   EI     = (const int*)  d_in[1];
  const float* W1     = (const float*)d_in[2];
  const float* asrc1  = (const float*)d_in[3];
  const float* adst1  = (const float*)d_in[4];
  const float* b1     = (const float*)d_in[5];
  const float* W2     = (const float*)d_in[6];
  const float* asrc2  = (const float*)d_in[7];
  const float* adst2  = (const float*)d_in[8];
  const float* b2     = (const float*)d_in[9];

  const int Fhid = in_sizes[5];            // 128
  const int Fout = in_sizes[9];            // 64
  const int Fin  = in_sizes[2] / Fhid;     // 128
  const int N    = in_sizes[0] / Fin;      // 50000
  const int E    = in_sizes[1] / 2;        // 800000
  const int Etot = E + N;

  // Workspace layout (floats)
  float*    h1    = (float*)d_ws;                     // N*Fhid
  float*    agg1  = h1 + (size_t)N * Fhid;            // N*Fhid (becomes x2)
  float*    h2    = agg1 + (size_t)N * Fhid;          // N*Fout
  float*    as_   = h2 + (size_t)N * Fout;            // N
  float*    ad_   = as_ + N;                          // N
  unsigned* emax  = (unsigned*)(ad_ + N);             // N
  float*    denom = (float*)(emax + N);               // N
  float*    ew    = denom + N;                        // Etot
  float*    out   = (float*)d_out;                    // N*Fout

  const int mt   = (N + 15) / 16;                     // GEMM M tiles
  const int egrd = (Etot + 255) / 256;                // thread-per-edge grids
  const int wgrd = (Etot + 7) / 8;                    // wave-per-edge grids (8 waves/block)

  // ---------------- Layer 1 ----------------
  gat_gemm_f32_wmma<<<mt, 32 * (Fhid / 16), 0, stream>>>(X, W1, h1, N, Fin, Fhid);
  gat_alphas<<<(N + 7) / 8, 256, 0, stream>>>(h1, asrc1, adst1, as_, ad_, N, Fhid);
  {
    long long tot = (long long)N * Fhid;
    gat_init<<<(int)((tot + 255) / 256), 256, 0, stream>>>(emax, denom, agg1, N, Fhid);
  }
  gat_edge_logits<<<egrd, 256, 0, stream>>>(EI, as_, ad_, ew, emax, E, N);
  gat_edge_expsum<<<egrd, 256, 0, stream>>>(EI, ew, emax, denom, E, N);
  gat_edge_scatter<<<wgrd, 256, 0, stream>>>(EI, ew, denom, h1, agg1, E, N, Fhid);
  {
    long long tot = (long long)N * Fhid;
    gat_finalize<<<(int)((tot + 255) / 256), 256, 0, stream>>>(agg1, b1, N, Fhid, 1);
  }

  // ---------------- Layer 2 ----------------
  gat_gemm_f32_wmma<<<mt, 32 * (Fout / 16), 0, stream>>>(agg1, W2, h2, N, Fhid, Fout);
  gat_alphas<<<(N + 7) / 8, 256, 0, stream>>>(h2, asrc2, adst2, as_, ad_, N, Fout);
  {
    long long tot = (long long)N * Fout;
    gat_init<<<(int)((tot + 255) / 256), 256, 0, stream>>>(emax, denom, out, N, Fout);
  }
  gat_edge_logits<<<egrd, 256, 0, stream>>>(EI, as_, ad_, ew, emax, E, N);
  gat_edge_expsum<<<egrd, 256, 0, stream>>>(EI, ew, emax, denom, E, N);
  gat_edge_scatter<<<wgrd, 256, 0, stream>>>(EI, ew, denom, h2, out, E, N, Fout);
  {
    long long tot = (long long)N * Fout;
    gat_finalize<<<(int)((tot + 255) / 256), 256, 0, stream>>>(out, b2, N, Fout, 0);
  }
}